// KernelConv2d_661424964036
// MI455X (gfx1250) — compile-verified
//
#include <hip/hip_runtime.h>

typedef __attribute__((ext_vector_type(2))) float v2f;
typedef __attribute__((ext_vector_type(8))) float v8f;

#define WW   256
#define HWSZ 65536
#define CCH  8
#define KKT  25
#define NEG_SLOPE 0.2f

// One wave32 handles 16 consecutive pixels (same n, ch, row) as a 16x16 WMMA
// whose diagonal is the 16 per-pixel length-25 dot products.
//   A (16x4 f32):  lane m = lane&15, VGPR j holds K = 2*(lane>>4)+j  (patch)
//   B (4x16 f32):  col n = lane&15, VGPR j holds row K = 2*(lane>>4)+j (weights)
//   D diag: p<8 -> lane p, acc[p];  p>=8 -> lane p+16, acc[p-8]

__device__ __forceinline__ void mma_extract_store(const v2f A[7], const v2f B[7],
                                                  int lane, float* __restrict__ outp)
{
    v8f acc = {0.f, 0.f, 0.f, 0.f, 0.f, 0.f, 0.f, 0.f};
#pragma unroll
    for (int kc = 0; kc < 7; ++kc) {
        acc = __builtin_amdgcn_wmma_f32_16x16x4_f32(
            /*neg_a=*/false, A[kc], /*neg_b=*/false, B[kc],
            /*c_mod=*/(short)0, acc, /*reuse_a=*/false, /*reuse_b=*/false);
    }
    const int idx = lane & 7;
    float v = acc[0];
#pragma unroll
    for (int i = 1; i < 8; ++i) v = (idx == i) ? acc[i] : v;
    v = (v >= 0.0f) ? v : NEG_SLOPE * v;   // leaky_relu(0.2)
    if (lane < 8 || lane >= 24) {
        const int p = (lane < 16) ? lane : (lane - 16);
        __builtin_nontemporal_store(v, outp + p);
    }
}

// grid = (2, 256, 32): blockIdx.x*8+wave = x-tile, blockIdx.y = row, blockIdx.z = plane
__global__ __launch_bounds__(256) void dynconv5x5_wmma(
    const float* __restrict__ x,     // (4, 8, 256, 256)
    const float* __restrict__ kern,  // (4, 8*25, 256, 256)
    float* __restrict__ out)         // (4, 8, 256, 256)
{
    const int lane  = threadIdx.x & 31;
    const int wave  = threadIdx.x >> 5;
    const int tx    = (blockIdx.x << 3) + wave;   // 0..15
    const int y     = blockIdx.y;                 // 0..255
    const int plane = blockIdx.z;                 // 0..31 (n*8+ch)

    const int x0 = tx << 4;
    const int hf = lane >> 4;          // half-wave selector (K sub-pair)
    const int lp = lane & 15;          // A row (pixel) == B column (pixel)

    const float* __restrict__ xp = x + plane * HWSZ;
    const int kbase = (plane * KKT) * HWSZ + y * WW + x0 + lp;
    float* __restrict__ outp = out + plane * HWSZ + y * WW + x0;

    // Wave-uniform interior test: 5x5 window never leaves the image.
    if ((y >= 2) & (y <= 253) & (tx >= 1) & (tx <= 14)) {
        // ---- FAST PATH (~86% of waves) ----
        // Lane tap t = t0 + 2*hf, pixel-relative offset OFF(t) = (t/5)*256 + t%5
        // (patch origin (-2,-2) folded into the base).  OFF(t0+2)-OFF(t0) in {2,253}.
        // 3 per-lane pointers; every load is then pointer + compile-time ioffset.
        const float* __restrict__ xb    = xp + (y - 2) * WW + (x0 - 2);
        const float* __restrict__ pA2   = xb + lp + hf * 2;
        const float* __restrict__ pA253 = xb + lp + hf * 253;
        const float* __restrict__ pB    = kern + kbase + hf * (2 * HWSZ);

        constexpr int T0[12]   = {0, 1, 4,   5,   8,   9,   12,  13,  16,  17,  20,   21};
        constexpr int OFF0[12] = {0, 1, 4, 256, 259, 260, 514, 515, 769, 770, 1024, 1025};
        constexpr int D253[12] = {0, 0, 1,   0,   1,   1,   0,   1,   0,   0,   0,    0};

        v2f A[7], B[7];
#pragma unroll
        for (int i = 0; i < 12; ++i) {
            const int kc = T0[i] >> 2;
            const int j  = T0[i] & 3;
            A[kc][j] = D253[i] ? pA253[OFF0[i]] : pA2[OFF0[i]];
            B[kc][j] = __builtin_nontemporal_load(pB + T0[i] * HWSZ);
        }
        // chunk 6: only tap 24 is real (hf==0, j==0); taps 25..27 statically zero
        const float a24 = xb[lp + 1028];                                  // OFF(24)
        const float b24 = __builtin_nontemporal_load(kern + kbase + 24 * HWSZ);
        A[6][0] = hf ? 0.0f : a24;
        B[6][0] = hf ? 0.0f : b24;
        A[6][1] = 0.0f;
        B[6][1] = 0.0f;

        mma_extract_store(A, B, lane, outp);   // own WMMA chain: not mergeable
        return;
    }

    // ---- GENERIC PATH (edge tiles): replicate padding via clamps ----
    {
        v2f A[7], B[7];
#pragma unroll
        for (int kc = 0; kc < 7; ++kc) {
#pragma unroll
            for (int j = 0; j < 2; ++j) {
                const int t  = 4 * kc + 2 * hf + j;        // tap index 0..27
                const int tc = (t < KKT) ? t : (KKT - 1);  // clamp for address
                const int dy = tc / 5 - 2;
                const int dx = tc % 5 - 2;
                int cy = y + dy;       cy = cy < 0 ? 0 : (cy > 255 ? 255 : cy);
                int cx = x0 + lp + dx; cx = cx < 0 ? 0 : (cx > 255 ? 255 : cx);
                const float av = xp[cy * WW + cx];
                const float bv = __builtin_nontemporal_load(kern + kbase + tc * HWSZ);
                const bool valid = (t < KKT);
                A[kc][j] = valid ? av : 0.0f;
                B[kc][j] = valid ? bv : 0.0f;
            }
        }
        mma_extract_store(A, B, lane, outp);
    }
}

extern "C" void kernel_launch(void* const* d_in, const int* in_sizes, int n_in,
                              void* d_out, int out_size, void* d_ws, size_t ws_size,
                              hipStream_t stream) {
    (void)in_sizes; (void)n_in; (void)d_ws; (void)ws_size; (void)out_size;
    const float* x    = (const float*)d_in[0];   // (4,8,256,256) f32
    const float* kern = (const float*)d_in[1];   // (4,200,256,256) f32
    float*       out  = (float*)d_out;           // (4,8,256,256) f32

    // (x-tile groups of 8, rows, planes); 8 waves per 256-thread block -> 131072 waves
    dim3 grid(2, 256, 32);
    dynconv5x5_wmma<<<grid, 256, 0, stream>>>(x, kern, out);
}